// PaddingBottleneck_2723009265989
// MI455X (gfx1250) — compile-verified
//
#include <hip/hip_runtime.h>

// MI455X / gfx1250, wave32.
// Reference: B=16, T=2048; x: (16, 2049, 2048) f32.
//   sig_norm[b,t] = sum_d |x[b,1+d,t]|         (column sums)
//   w = sigmoid(x[b,0,:] - sig_norm)
//   out[b,0,:]   = w * x[b,0,:]
//   out[b,1+d,:] = (1 - w[b,d]) * x[b,1+d,:]
// Memory-bound: ~805 MB total traffic -> ~35 us at 23.3 TB/s.
// Column reduction done on the matrix pipe via V_WMMA_F32_16X16X4_F32 with an
// all-ones A matrix (D = ones(16x4) x |B|(4x16) + C => every D row = colsums).

typedef __attribute__((ext_vector_type(2))) float v2f;
typedef __attribute__((ext_vector_type(4))) float v4f;
typedef __attribute__((ext_vector_type(8))) float v8f;

static constexpr int kB = 16;
static constexpr int kT = 2048;        // columns, also number of sig rows
static constexpr int kRows = 2049;     // x rows per batch (pad + sig)

// ---------------------------------------------------------------------------
// Kernel 1: per-wave WMMA column reduction over 16 columns x 2048 rows.
// B-matrix (4x16 f32, 2 VGPRs) layout per ISA 7.12.2:
//   VGPR0: lanes 0-15 -> K=0 row, lanes 16-31 -> K=2 row
//   VGPR1: lanes 0-15 -> K=1 row, lanes 16-31 -> K=3 row
// Each lane therefore loads rows (d+row_off) and (d+row_off+1) at its column,
// where row_off = 0 for lanes 0-15 and 2 for lanes 16-31.
// ---------------------------------------------------------------------------
__global__ __launch_bounds__(256)
void pb_colnorm_wmma(const float* __restrict__ x,
                     float* __restrict__ out,
                     float* __restrict__ wbuf) {
  const int lane   = threadIdx.x & 31;
  const int wave   = blockIdx.x * 8 + (threadIdx.x >> 5);   // 0..2047
  const int b      = wave >> 7;                             // 128 tiles / batch
  const int t0     = (wave & 127) << 4;                     // 16 cols per wave
  const int n      = lane & 15;
  const int rowoff = (lane >> 4) << 1;                      // 0 or 2

  const float* sig = x + (size_t)b * kRows * kT + kT;       // row d=0
  const float* p   = sig + (size_t)rowoff * kT + (t0 + n);

  v2f a; a[0] = 1.0f; a[1] = 1.0f;                          // ones A matrix
  v8f c0 = {}, c1 = {}, c2 = {}, c3 = {};                   // 4 accum chains

  for (int d = 0; d < kT; d += 16) {
    const float* q = p + (size_t)d * kT;
    v2f b0, b1, b2, b3;
    b0[0] = __builtin_fabsf(q[0 * kT]);  b0[1] = __builtin_fabsf(q[1 * kT]);
    b1[0] = __builtin_fabsf(q[4 * kT]);  b1[1] = __builtin_fabsf(q[5 * kT]);
    b2[0] = __builtin_fabsf(q[8 * kT]);  b2[1] = __builtin_fabsf(q[9 * kT]);
    b3[0] = __builtin_fabsf(q[12 * kT]); b3[1] = __builtin_fabsf(q[13 * kT]);
    c0 = __builtin_amdgcn_wmma_f32_16x16x4_f32(false, a, false, b0, (short)0, c0, false, false);
    c1 = __builtin_amdgcn_wmma_f32_16x16x4_f32(false, a, false, b1, (short)0, c1, false, false);
    c2 = __builtin_amdgcn_wmma_f32_16x16x4_f32(false, a, false, b2, (short)0, c2, false, false);
    c3 = __builtin_amdgcn_wmma_f32_16x16x4_f32(false, a, false, b3, (short)0, c3, false, false);
  }

  // Every D row holds the column sums; element 0 of the accumulator in lanes
  // 0-15 is row M=0, cols t0..t0+15.
  const float norm = (c0[0] + c1[0]) + (c2[0] + c3[0]);

  if (lane < 16) {
    const int t    = t0 + n;
    const size_t o = (size_t)b * kRows * kT + t;   // pad row / out row 0
    const float pv = x[o];
    const float wv = 1.0f / (1.0f + __expf(-(pv - norm)));
    out[o]              = wv * pv;
    wbuf[b * kT + t]    = wv;                      // stash w for pass 2
  }
}

// ---------------------------------------------------------------------------
// Kernel 2: streaming scale, out[b,1+d,t] = (1 - w[b,d]) * sig[b,d,t].
// float4 per lane; NT stores keep the output stream out of L2 so pass-1's
// resident sig lines survive for the re-read.
// ---------------------------------------------------------------------------
__global__ __launch_bounds__(256)
void pb_scale(const float* __restrict__ x,
              const float* __restrict__ wbuf,
              float* __restrict__ out) {
  const size_t idx  = (size_t)blockIdx.x * blockDim.x + threadIdx.x; // float4 id
  const size_t col4 = idx & 511;            // 512 float4 per 2048-col row
  const size_t row  = idx >> 9;             // b*2048 + d
  const size_t b    = row >> 11;

  const float s = 1.0f - wbuf[row];
  // x/out element row (b*2049 + 1 + d) == row + b + 1
  const size_t off4 = (row + b + 1) * 512 + col4;

  v4f v = ((const v4f*)x)[off4];
  v *= s;
  __builtin_nontemporal_store(v, ((v4f*)out) + off4);
}

extern "C" void kernel_launch(void* const* d_in, const int* in_sizes, int n_in,
                              void* d_out, int out_size, void* d_ws, size_t ws_size,
                              hipStream_t stream) {
  const float* x  = (const float*)d_in[0];
  float* out      = (float*)d_out;
  float* wbuf     = (float*)d_ws;   // needs 16*2048*4 = 128 KB of scratch

  // Pass 1: 2048 waves = 256 blocks x 8 waves; each wave -> 16 columns.
  pb_colnorm_wmma<<<256, 256, 0, stream>>>(x, out, wbuf);

  // Pass 2: 16*2048*512 float4 elements / 256 threads = 65536 blocks.
  pb_scale<<<65536, 256, 0, stream>>>(x, wbuf, out);
}